// Get3DLUT_identity_21191368639341
// MI455X (gfx1250) — compile-verified
//
#include <hip/hip_runtime.h>

// Trilinear 3D LUT apply for MI455X (gfx1250).
// Roofline: read x (100.7 MB) + write out (100.7 MB) => ~8.6 us @ 23.3 TB/s.
// LUT repacked to interleaved float4 (575 KB) and kept L2-resident (192 MB L2);
// image stream uses NT temporal hints so it never evicts the LUT.
// Per pixel: 8x global_load_b128 gathers + 7 v4f lerps (v_pk_*_f32 eligible).

typedef __attribute__((ext_vector_type(4))) float v4f;

#define LUT_DIM   33
#define LUT_DIM2  (LUT_DIM * LUT_DIM)            // 1089
#define NCELL     (LUT_DIM * LUT_DIM * LUT_DIM)  // 35937
#define IMG_HW    (1024 * 1024)
#define NBATCH    8
#define NPIX      (NBATCH * IMG_HW)              // 8388608
#define PIX_PER_THREAD 4

// 1 / float32(1.000001 / 32)
#define INV_BINSIZE 31.99996800003200f

__device__ __forceinline__ v4f lerp4(v4f a, v4f b, float w) { return a + w * (b - a); }
__device__ __forceinline__ float lerp1(float a, float b, float w) { return a + w * (b - a); }

__device__ __forceinline__ void cell_coords(float r, float g, float b,
                                            int& base, float& rd, float& gd, float& bd) {
    float tr = r * INV_BINSIZE, tg = g * INV_BINSIZE, tb = b * INV_BINSIZE;
    // clamp lowers to v_med3_f32
    float fr = fminf(fmaxf(floorf(tr), 0.0f), (float)(LUT_DIM - 2));
    float fg = fminf(fmaxf(floorf(tg), 0.0f), (float)(LUT_DIM - 2));
    float fb = fminf(fmaxf(floorf(tb), 0.0f), (float)(LUT_DIM - 2));
    rd = tr - fr; gd = tg - fg; bd = tb - fb;
    base = ((int)fb * LUT_DIM + (int)fg) * LUT_DIM + (int)fr;  // 2x v_mad
}

// Packed path: 8 b128 gathers with immediate offsets off one base address,
// nested lerp chain (algebraically == reference's 8-corner weight sum).
__device__ __forceinline__ v4f sample_packed(const v4f* __restrict__ lut4,
                                             float r, float g, float b) {
    int base; float rd, gd, bd;
    cell_coords(r, g, b, base, rd, gd, bd);
    v4f c000 = lut4[base];
    v4f c001 = lut4[base + 1];
    v4f c010 = lut4[base + LUT_DIM];
    v4f c011 = lut4[base + LUT_DIM + 1];
    v4f c100 = lut4[base + LUT_DIM2];
    v4f c101 = lut4[base + LUT_DIM2 + 1];
    v4f c110 = lut4[base + LUT_DIM2 + LUT_DIM];
    v4f c111 = lut4[base + LUT_DIM2 + LUT_DIM + 1];
    v4f c0 = lerp4(lerp4(c000, c001, rd), lerp4(c010, c011, rd), gd);
    v4f c1 = lerp4(lerp4(c100, c101, rd), lerp4(c110, c111, rd), gd);
    return lerp4(c0, c1, bd);
}

// Fallback path (scratch too small): planar scalar gathers.
__device__ __forceinline__ void sample_planar(const float* __restrict__ lut,
                                              float r, float g, float b, float* o) {
    int base; float rd, gd, bd;
    cell_coords(r, g, b, base, rd, gd, bd);
#pragma unroll
    for (int c = 0; c < 3; ++c) {
        const float* L = lut + c * NCELL;
        float a00 = L[base],             a01 = L[base + 1];
        float a10 = L[base + LUT_DIM],   a11 = L[base + LUT_DIM + 1];
        float b00 = L[base + LUT_DIM2],            b01 = L[base + LUT_DIM2 + 1];
        float b10 = L[base + LUT_DIM2 + LUT_DIM],  b11 = L[base + LUT_DIM2 + LUT_DIM + 1];
        float v0 = lerp1(lerp1(a00, a01, rd), lerp1(a10, a11, rd), gd);
        float v1 = lerp1(lerp1(b00, b01, rd), lerp1(b10, b11, rd), gd);
        o[c] = lerp1(v0, v1, bd);
    }
}

// ---------------------------------------------------------------------------
// Main kernel (defined first so its asm leads the .text section).
// 4 pixels/thread: coalesced b128 NT loads of the 3 planes, LUT gathers from
// L2, coalesced b128 NT stores. Block 0 warms the LUT via global_prefetch_b8.
// ---------------------------------------------------------------------------
template <bool PACKED>
__global__ __launch_bounds__(256)
void lut3d_apply_kernel(const float* __restrict__ x,
                        const float* __restrict__ lutp,
                        const v4f*  __restrict__ lut4,
                        float* __restrict__ out) {
    if (blockIdx.x == 0) {
        const char* pb = PACKED ? (const char*)lut4 : (const char*)lutp;
        const int nbytes = PACKED ? NCELL * 16 : NCELL * 12;
        for (int off = (int)threadIdx.x * 256; off < nbytes; off += 256 * 256)
            __builtin_prefetch(pb + off, 0, 3);
    }

    unsigned int t  = blockIdx.x * blockDim.x + threadIdx.x;
    unsigned int p0 = t * PIX_PER_THREAD;           // < 2^23, fits easily
    if (p0 >= NPIX) return;

    // 4 consecutive pixels never straddle a batch image (HW % 4 == 0).
    unsigned int img = p0 >> 20;                     // / IMG_HW
    unsigned int pix = p0 & (IMG_HW - 1);
    unsigned int off = img * (3u * IMG_HW) + pix;    // shared by x and out

    const float* in = x + off;
    v4f r4 = __builtin_nontemporal_load((const v4f*)(in));
    v4f g4 = __builtin_nontemporal_load((const v4f*)(in + IMG_HW));
    v4f b4 = __builtin_nontemporal_load((const v4f*)(in + 2 * IMG_HW));

    v4f orv, ogv, obv;
#pragma unroll
    for (int k = 0; k < PIX_PER_THREAD; ++k) {
        if (PACKED) {
            v4f c = sample_packed(lut4, r4[k], g4[k], b4[k]);
            orv[k] = c.x; ogv[k] = c.y; obv[k] = c.z;
        } else {
            float o[3];
            sample_planar(lutp, r4[k], g4[k], b4[k], o);
            orv[k] = o[0]; ogv[k] = o[1]; obv[k] = o[2];
        }
    }

    float* op = out + off;
    __builtin_nontemporal_store(orv, (v4f*)(op));
    __builtin_nontemporal_store(ogv, (v4f*)(op + IMG_HW));
    __builtin_nontemporal_store(obv, (v4f*)(op + 2 * IMG_HW));
}

template __global__ void lut3d_apply_kernel<true>(const float*, const float*, const v4f*, float*);
template __global__ void lut3d_apply_kernel<false>(const float*, const float*, const v4f*, float*);

// ---------------------------------------------------------------------------
// Repack planar (3, 33^3) LUT into interleaved float4 {r,g,b,0} in d_ws.
// Side effect: pulls the whole LUT through L2, pre-warming it.
// ---------------------------------------------------------------------------
__global__ __launch_bounds__(256)
void repack_lut_kernel(const float* __restrict__ lut, v4f* __restrict__ lut4) {
    int i = blockIdx.x * blockDim.x + threadIdx.x;
    if (i < NCELL) {
        v4f c;
        c.x = lut[i];
        c.y = lut[NCELL + i];
        c.z = lut[2 * NCELL + i];
        c.w = 0.0f;
        lut4[i] = c;
    }
}

extern "C" void kernel_launch(void* const* d_in, const int* in_sizes, int n_in,
                              void* d_out, int out_size, void* d_ws, size_t ws_size,
                              hipStream_t stream) {
    const float* lut = (const float*)d_in[0]; // (3, 33, 33, 33) fp32
    const float* x   = (const float*)d_in[1]; // (8, 3, 1024, 1024) fp32
    float* out = (float*)d_out;

    const int nthreads = NPIX / PIX_PER_THREAD; // 2,097,152
    dim3 block(256);
    dim3 grid(nthreads / 256);                  // 8192 blocks, exact

    if (ws_size >= (size_t)NCELL * sizeof(v4f)) {
        v4f* lut4 = (v4f*)d_ws;
        repack_lut_kernel<<<(NCELL + 255) / 256, 256, 0, stream>>>(lut, lut4);
        lut3d_apply_kernel<true><<<grid, block, 0, stream>>>(x, lut, lut4, out);
    } else {
        lut3d_apply_kernel<false><<<grid, block, 0, stream>>>(x, lut, nullptr, out);
    }
}